// word2vec_neg_sampling_29798483100076
// MI455X (gfx1250) — compile-verified
//
#include <hip/hip_runtime.h>
#include <math.h>

// word2vec SGNS loss on MI455X (gfx1250, wave32).
// Memory-bound gather problem (~100MB, ~4.4us @ 23.3TB/s HBM; tables fit in 192MB L2).
// Dot products are mapped onto V_WMMA_F32_16X16X4_F32 chains (K=128 -> 32 wmma/chain),
// taking the 16x16 product's diagonal as the 16 per-batch-element scores.

typedef __attribute__((ext_vector_type(2))) float v2f;
typedef __attribute__((ext_vector_type(8))) float v8f;

#define EMBED 128
#define BATCH 16384
#define NEG   10
#define TILES (BATCH / 16)          // 1024 tiles of 16 batch elements
#define WAVES_PER_BLOCK 4
#define BLOCKS (TILES / WAVES_PER_BLOCK)

__device__ __forceinline__ float log_sigmoid(float x) {
    // numerically stable: min(x,0) - log1p(exp(-|x|))
    return fminf(x, 0.0f) - log1pf(expf(-fabsf(x)));
}

__global__ __launch_bounds__(WAVES_PER_BLOCK * 32)
void w2v_tile_kernel(const int* __restrict__ IW,    // input_word   [B]
                     const int* __restrict__ CW,    // context_word [B]
                     const int* __restrict__ NS,    // neg_samples  [B, NEG]
                     const float* __restrict__ Win, // W_in  [V, E]
                     const float* __restrict__ Wctx,// W_ctx [V, E]
                     float* __restrict__ partial)   // [TILES]
{
    __shared__ float cbuf[WAVES_PER_BLOCK][16][16];

    const int lane = threadIdx.x & 31;
    const int wave = threadIdx.x >> 5;
    const int tile = blockIdx.x * WAVES_PER_BLOCK + wave;
    const int base = tile * 16;
    const int col  = lane & 15;   // M (A row) / N (B col) owned by this lane
    const int hi   = lane >> 4;   // lane-half selects K sub-pair per ISA layout

    // ---- A operand: 16 gathered W_in rows, resident across all 11 chains ----
    // ISA 32-bit A 16x4 layout: lane L holds row M=L%15.. (L&15), K = 4s + 2*hi + {0,1}
    const int iw = IW[base + col];
    const float* arow = Win + (size_t)iw * EMBED + 2 * hi;
    v2f af[32];
#pragma unroll
    for (int s = 0; s < 32; ++s)
        af[s] = *(const v2f*)(arow + 4 * s);

    float acc = 0.0f;   // lanes 0..15 accumulate log-sig terms for element `col`

#pragma unroll
    for (int t = 0; t < 1 + NEG; ++t) {
        // B operand column N=col is the context (t==0) or t-1'th negative row.
        const int ridx = (t == 0) ? CW[base + col]
                                  : NS[(base + col) * NEG + (t - 1)];
        const float* brow = Wctx + (size_t)ridx * EMBED + 2 * hi;
        v2f bf[32];
#pragma unroll
        for (int s = 0; s < 32; ++s)
            bf[s] = *(const v2f*)(brow + 4 * s);

        // K-chain: C[m][n] = sum_k emb_in[m][k] * emb_row_n[k]
        v8f c = {};
#pragma unroll
        for (int s = 0; s < 32; ++s)
            c = __builtin_amdgcn_wmma_f32_16x16x4_f32(
                    false, af[s], false, bf[s], (short)0, c, false, false);

        // Stage C into LDS (C/D layout: VGPR r, lane-half hi -> M = r + 8*hi, N = lane&15)
        float* cb = &cbuf[wave][0][0];
#pragma unroll
        for (int r = 0; r < 8; ++r)
            cb[(r + 8 * hi) * 16 + col] = c[r];
        __syncthreads();                 // order LDS store->load (uniform flow)

        const float diag = cb[col * 16 + col];   // score for batch element `col`
        const float x = (t == 0) ? diag : -diag; // neg_score = -dot
        acc += log_sigmoid(x);
        __syncthreads();                 // protect cbuf before next chain's stores
    }

    if (hi) acc = 0.0f;                  // high half duplicated the work; drop it
#pragma unroll
    for (int off = 8; off >= 1; off >>= 1)
        acc += __shfl_xor(acc, off, 32); // sum over 16 low lanes
    if (lane == 0)
        partial[tile] = acc;
}

__global__ __launch_bounds__(256)
void w2v_reduce_kernel(const float* __restrict__ partial, float* __restrict__ out) {
    __shared__ float sbuf[256];
    float s = 0.0f;
    for (int i = threadIdx.x; i < TILES; i += 256)
        s += partial[i];
    sbuf[threadIdx.x] = s;
    __syncthreads();
#pragma unroll
    for (int stride = 128; stride >= 1; stride >>= 1) {
        if (threadIdx.x < stride) sbuf[threadIdx.x] += sbuf[threadIdx.x + stride];
        __syncthreads();
    }
    if (threadIdx.x == 0)
        out[0] = -sbuf[0] / (float)BATCH;   // -mean(out_loss + noise_loss)
}

extern "C" void kernel_launch(void* const* d_in, const int* in_sizes, int n_in,
                              void* d_out, int out_size, void* d_ws, size_t ws_size,
                              hipStream_t stream) {
    const int*   IW   = (const int*)d_in[0];
    const int*   CW   = (const int*)d_in[1];
    const int*   NS   = (const int*)d_in[2];
    const float* Win  = (const float*)d_in[3];
    const float* Wctx = (const float*)d_in[4];
    float* out     = (float*)d_out;
    float* partial = (float*)d_ws;   // 1024 floats, fully rewritten every call

    w2v_tile_kernel<<<BLOCKS, WAVES_PER_BLOCK * 32, 0, stream>>>(
        IW, CW, NS, Win, Wctx, partial);
    w2v_reduce_kernel<<<1, 256, 0, stream>>>(partial, out);
}